// ScaledDotProductAttention_40690520162940
// MI455X (gfx1250) — compile-verified
//
#include <hip/hip_runtime.h>

// Fused attention for B=8, S=2048, DK=DV=128 (fp32 in, fp32 out).
// Outputs (concatenated in d_out): O [B,S,DV] then attention weights [B,S,S].
//
// One workgroup (256 threads = 8 wave32) per (batch, 32-query tile): 512 blocks.
// Score panel 32x2048 f32 (~257KB padded) lives in LDS - only possible because
// CDNA5 gives 320KB per WGP. 32-row tiles + shared-B register reuse cut K/V
// L2 traffic 2x vs a 16-row design (K and V each read exactly once per block).
// Weights are streamed to HBM with non-temporal stores so the 134MB write
// stream does not evict the L2-resident K/V working set (16MB total).

#define S_LEN   2048
#define DK_DIM  128
#define DV_DIM  128
#define BATCH   8
#define QROWS   32              // query rows per block
#define NKT     (S_LEN / 16)    // 128 key tiles per row panel
#define NWAVES  8
#define SC_STRIDE 2052          // 2048+4: column accesses hit 16 distinct banks
#define LQ_STRIDE 132           // 128+4: same trick for the Q tile

typedef float v2f __attribute__((ext_vector_type(2)));
typedef float v8f __attribute__((ext_vector_type(8)));

__global__ __launch_bounds__(256)
void attn_fused_f32wmma(const float* __restrict__ Q,
                        const float* __restrict__ K,
                        const float* __restrict__ V,
                        float* __restrict__ O,
                        float* __restrict__ W)
{
    extern __shared__ float smem[];
    float* lq = smem;                         // [QROWS][LQ_STRIDE]
    float* sc = smem + QROWS * LQ_STRIDE;     // [QROWS][SC_STRIDE]

    const int tid  = threadIdx.x;
    const int wave = tid >> 5;
    const int lane = tid & 31;
    const int lo   = lane & 15;   // M / N index within 16
    const int hi   = lane >> 4;   // half-wave select

    const int b  = blockIdx.x >> 6;    // / 64
    const int qt = blockIdx.x & 63;    // 32-query tile index

    const float* Qb = Q + ((size_t)b * S_LEN + (size_t)qt * QROWS) * DK_DIM;
    const float* Kb = K + (size_t)b * S_LEN * DK_DIM;
    const float* Vb = V + (size_t)b * S_LEN * DV_DIM;

    // ---- Stage Q tile (32x128) into LDS, fully coalesced ----
    for (int idx = tid; idx < QROWS * DK_DIM; idx += 256) {
        int r = idx >> 7;       // / 128
        int c = idx & 127;
        lq[r * LQ_STRIDE + c] = Qb[idx];
    }
    __syncthreads();

    // ---- Phase 1: scores = Q * K^T, 32 x 2048 ----
    // Each wave owns 16 of the 128 key tiles; the SAME K float2 feeds both
    // 16-row halves of the query tile (two accumulators, B reused).
    // A (16x4 f32): a[i] = A[lo][2*hi + i]           -> contiguous float2
    // B (4x16 f32): b[i] = B[i + 2*hi][lo] = K[key=lo][kk + 2*hi + i]
    for (int jt = wave; jt < NKT; jt += NWAVES) {
        v8f acc0 = {};
        v8f acc1 = {};
        const float* q0   = lq + (size_t)lo * LQ_STRIDE + 2 * hi;
        const float* q1   = lq + (size_t)(16 + lo) * LQ_STRIDE + 2 * hi;
        const float* krow = Kb + ((size_t)(jt * 16 + lo)) * DK_DIM + 2 * hi;
        #pragma unroll 8
        for (int kk = 0; kk < DK_DIM; kk += 4) {
            v2f bm = *(const v2f*)(krow + kk);   // loaded once, used twice
            v2f a0 = *(const v2f*)(q0 + kk);
            v2f a1 = *(const v2f*)(q1 + kk);
            acc0 = __builtin_amdgcn_wmma_f32_16x16x4_f32(
                false, a0, false, bm, (short)0, acc0, false, false);
            acc1 = __builtin_amdgcn_wmma_f32_16x16x4_f32(
                false, a1, false, bm, (short)0, acc1, false, false);
        }
        // C/D layout: c[i] = C[i + 8*hi][lo]
        float* d0 = sc + (size_t)(8 * hi) * SC_STRIDE + jt * 16 + lo;
        float* d1 = sc + (size_t)(16 + 8 * hi) * SC_STRIDE + jt * 16 + lo;
        #pragma unroll
        for (int i = 0; i < 8; ++i) d0[(size_t)i * SC_STRIDE] = acc0[i];
        #pragma unroll
        for (int i = 0; i < 8; ++i) d1[(size_t)i * SC_STRIDE] = acc1[i];
    }
    __syncthreads();

    // ---- Phase 2: row softmax (4 rows per wave), weights written once (NT) ----
    float* Wbase = W + ((size_t)b * S_LEN + (size_t)qt * QROWS) * S_LEN;
    for (int r = wave * 4; r < wave * 4 + 4; ++r) {
        float* row = sc + (size_t)r * SC_STRIDE;
        float m = -3.402823466e+38f;
        for (int c = lane; c < S_LEN; c += 32) m = fmaxf(m, row[c]);
        #pragma unroll
        for (int off = 16; off > 0; off >>= 1)
            m = fmaxf(m, __shfl_xor(m, off, 32));
        float sum = 0.0f;
        for (int c = lane; c < S_LEN; c += 32) {
            float e = __expf(row[c] - m);
            row[c] = e;
            sum += e;
        }
        #pragma unroll
        for (int off = 16; off > 0; off >>= 1)
            sum += __shfl_xor(sum, off, 32);
        const float inv = 1.0f / sum;
        float* wr = Wbase + (size_t)r * S_LEN;
        for (int c = lane; c < S_LEN; c += 32) {
            float wv = row[c] * inv;
            row[c] = wv;                              // normalized, for phase 3
            __builtin_nontemporal_store(wv, wr + c);  // stream 134MB past L2
        }
    }
    __syncthreads();

    // ---- Phase 3: O(32x128) = Wn(32x2048) @ V(2048x128) ----
    // Wave w owns output columns [16w, 16w+16); each V float is loaded ONCE
    // per block and feeds both 16-row halves (two accumulators).
    {
        v8f acc0 = {};
        v8f acc1 = {};
        const float* a0r  = sc + (size_t)lo * SC_STRIDE + 2 * hi;
        const float* a1r  = sc + (size_t)(16 + lo) * SC_STRIDE + 2 * hi;
        const float* vcol = Vb + wave * 16 + lo;
        #pragma unroll 4
        for (int k = 0; k < S_LEN; k += 4) {
            v2f bm;                                   // b[i] = V[k+2*hi+i][col]
            bm.x = vcol[(size_t)(k + 2 * hi)     * DV_DIM];
            bm.y = vcol[(size_t)(k + 2 * hi + 1) * DV_DIM];
            v2f a0 = *(const v2f*)(a0r + k);
            v2f a1 = *(const v2f*)(a1r + k);
            acc0 = __builtin_amdgcn_wmma_f32_16x16x4_f32(
                false, a0, false, bm, (short)0, acc0, false, false);
            acc1 = __builtin_amdgcn_wmma_f32_16x16x4_f32(
                false, a1, false, bm, (short)0, acc1, false, false);
        }
        float* o0 = O + ((size_t)b * S_LEN + (size_t)qt * QROWS + 8 * hi) * DV_DIM
                      + wave * 16 + lo;
        float* o1 = o0 + (size_t)16 * DV_DIM;
        #pragma unroll
        for (int i = 0; i < 8; ++i) o0[(size_t)i * DV_DIM] = acc0[i];
        #pragma unroll
        for (int i = 0; i < 8; ++i) o1[(size_t)i * DV_DIM] = acc1[i];
    }
}

extern "C" void kernel_launch(void* const* d_in, const int* in_sizes, int n_in,
                              void* d_out, int out_size, void* d_ws, size_t ws_size,
                              hipStream_t stream) {
    (void)in_sizes; (void)n_in; (void)out_size; (void)d_ws; (void)ws_size;
    const float* Q = (const float*)d_in[0];
    const float* K = (const float*)d_in[1];
    const float* V = (const float*)d_in[2];
    float* O = (float*)d_out;                                   // [B,S,DV]
    float* W = O + (size_t)BATCH * S_LEN * DV_DIM;              // [B,S,S]

    const size_t shmem =
        (size_t)(QROWS * LQ_STRIDE + QROWS * SC_STRIDE) * sizeof(float);
    hipFuncSetAttribute((const void*)attn_fused_f32wmma,
                        hipFuncAttributeMaxDynamicSharedMemorySize, (int)shmem);

    dim3 grid(BATCH * (S_LEN / QROWS));   // 512 workgroups
    dim3 block(256);                      // 8 wave32
    attn_fused_f32wmma<<<grid, block, shmem, stream>>>(Q, K, V, O, W);
}